// SpatialTransformer_2671469658636
// MI455X (gfx1250) — compile-verified
//
#include <hip/hip_runtime.h>
#include <math.h>

// Spatial transformer: affine grid sample, reflect padding, bilinear.
// x: (32, 3, 512, 512) f32, p: (32, 4) f32 [tx, ty, theta, scale]
// out: (32, 3, 512, 512) f32
//
// Memory-bound gather: ~100MB in + ~100MB out -> ~8.6us floor at 23.3 TB/s.
// Per-batch trig hoisted to a 32-thread precompute kernel. Hot kernel:
// index math + 6 b64 gathers (x-tap pairs fused) + 3 NT stores per thread.
// NT stores keep the write-once output from evicting the (100MB < 192MB L2)
// gathered input working set.

namespace {

constexpr int kB = 32;
constexpr int kC = 3;
constexpr int kH = 512;
constexpr int kW = 512;
constexpr int kHW = kH * kW;        // 262144
constexpr int kCHW = kC * kHW;      // per-batch elements

// two adjacent floats, 4-byte aligned (x0 may be odd)
struct __attribute__((aligned(4))) F2 { float x, y; };

__device__ __forceinline__ float reflect_coord(float coord, float size) {
    // matches jnp reference: fold |coord+0.5| across [0,size) with parity flip
    float c = fabsf(coord + 0.5f);
    float flips = floorf(c / size);
    float extra = c - flips * size;
    float half = flips * 0.5f;
    bool even = (floorf(half) == half);
    float r = (even ? extra : size - extra) - 0.5f;
    return fminf(fmaxf(r, 0.0f), size - 1.0f);
}

// Per-pixel body given the batch's affine coefficients.
// grid map:  gx = a*xs - d*ys + tx ;  gy = d*xs + a*ys + ty
__device__ __forceinline__ void sample_pixel(
    const float* __restrict__ x, float* __restrict__ out,
    int b, int pix, float a, float d, float tx, float ty) {
    const int ox = pix & (kW - 1);
    const int oy = pix >> 9;  // kW == 512

    const float xs = ((float)ox + 0.5f) * (2.0f / (float)kW) - 1.0f;
    const float ys = ((float)oy + 0.5f) * (2.0f / (float)kH) - 1.0f;

    const float gx = a * xs - d * ys + tx;
    const float gy = d * xs + a * ys + ty;

    float ix = ((gx + 1.0f) * (float)kW - 1.0f) * 0.5f;
    float iy = ((gy + 1.0f) * (float)kH - 1.0f) * 0.5f;
    ix = reflect_coord(ix, (float)kW);
    iy = reflect_coord(iy, (float)kH);

    const float ix0f = floorf(ix);
    const float iy0f = floorf(iy);
    const float fx = ix - ix0f;
    const float fy = iy - iy0f;

    // ix,iy already clipped to [0, size-1] => x0,y0 in [0, size-1]
    const int x0 = min((int)ix0f, kW - 1);
    const int y0 = min((int)iy0f, kH - 1);
    const int y1 = min(y0 + 1, kH - 1);

    // fuse the (x0, x0+1) tap pair into one b64 load at xb = min(x0, W-2);
    // when x0 == W-1 both taps are the same rightmost texel == pair.y
    const int xb = min(x0, kW - 2);
    const bool x_hi = (x0 > xb);  // x0 == W-1

    const float w00 = (1.0f - fx) * (1.0f - fy);
    const float w01 = fx * (1.0f - fy);
    const float w10 = (1.0f - fx) * fy;
    const float w11 = fx * fy;

    const int r0 = y0 * kW + xb;
    const int r1 = y1 * kW + xb;

    const size_t bbase = (size_t)b * kCHW;
    const float* __restrict__ img = x + bbase;
    float* __restrict__ op = out + bbase + (size_t)oy * kW + ox;

#pragma unroll
    for (int ch = 0; ch < kC; ++ch) {
        const float* __restrict__ im = img + ch * kHW;
        const F2 p0 = *(const F2*)(im + r0);
        const F2 p1 = *(const F2*)(im + r1);
        const float v00 = x_hi ? p0.y : p0.x;
        const float v01 = p0.y;
        const float v10 = x_hi ? p1.y : p1.x;
        const float v11 = p1.y;
        const float r = v00 * w00 + v01 * w01 + v10 * w10 + v11 * w11;
        // NT store: output is write-once; keep L2 for the gathered input
        __builtin_nontemporal_store(r, op + ch * kHW);
    }
}

// --- Pass 1: 32 lanes compute per-batch coefficients [a, d, tx, ty] --------
__global__ void precompute_coef_kernel(const float* __restrict__ p,
                                       float* __restrict__ coef) {
    const int b = threadIdx.x;
    if (b < kB) {
        const float tx = p[b * 4 + 0];
        const float ty = p[b * 4 + 1];
        const float th = p[b * 4 + 2];
        const float ts = p[b * 4 + 3];
        const float s = sinf(th);
        const float c = cosf(th);
        coef[b * 4 + 0] = ts * c;  // a  (== e)
        coef[b * 4 + 1] = ts * s;  // d  (== -b)
        coef[b * 4 + 2] = tx;
        coef[b * 4 + 3] = ty;
    }
}

// --- Pass 2: hot gather kernel, no transcendentals, no guard branch --------
__global__ __launch_bounds__(256) void spatial_transformer_kernel(
    const float* __restrict__ x, const float* __restrict__ coef,
    float* __restrict__ out) {
    const int b   = blockIdx.y;
    const int pix = blockIdx.x * blockDim.x + threadIdx.x;  // exact tiling

    // coef[b] is block-uniform -> s_load_b128
    const float a  = coef[b * 4 + 0];
    const float d  = coef[b * 4 + 1];
    const float tx = coef[b * 4 + 2];
    const float ty = coef[b * 4 + 3];

    sample_pixel(x, out, b, pix, a, d, tx, ty);
}

// --- Fallback: single kernel with in-kernel trig (if ws too small) ----------
__global__ __launch_bounds__(256) void spatial_transformer_fused_kernel(
    const float* __restrict__ x, const float* __restrict__ p,
    float* __restrict__ out) {
    const int b   = blockIdx.y;
    const int pix = blockIdx.x * blockDim.x + threadIdx.x;

    const float tx = p[b * 4 + 0];
    const float ty = p[b * 4 + 1];
    const float th = p[b * 4 + 2];
    const float ts = p[b * 4 + 3];
    const float a = ts * cosf(th);
    const float d = ts * sinf(th);

    sample_pixel(x, out, b, pix, a, d, tx, ty);
}

}  // namespace

extern "C" void kernel_launch(void* const* d_in, const int* in_sizes, int n_in,
                              void* d_out, int out_size, void* d_ws, size_t ws_size,
                              hipStream_t stream) {
    (void)in_sizes; (void)n_in; (void)out_size;
    const float* x = (const float*)d_in[0];
    const float* p = (const float*)d_in[1];
    float* out = (float*)d_out;

    dim3 block(256);                       // 8 wave32s
    dim3 grid(kHW / 256, kB);              // (1024, 32) -- exact tiling

    if (ws_size >= (size_t)(kB * 4 * sizeof(float))) {
        float* coef = (float*)d_ws;
        precompute_coef_kernel<<<1, 32, 0, stream>>>(p, coef);
        spatial_transformer_kernel<<<grid, block, 0, stream>>>(x, coef, out);
    } else {
        spatial_transformer_fused_kernel<<<grid, block, 0, stream>>>(x, p, out);
    }
}